// Predictor_5720896438560
// MI455X (gfx1250) — compile-verified
//
#include <hip/hip_runtime.h>
#include <hip/hip_bf16.h>

typedef unsigned long long u64;
typedef unsigned int u32;
typedef int v4i __attribute__((ext_vector_type(4)));

#define AS1 __attribute__((address_space(1)))
#define AS3 __attribute__((address_space(3)))

#define TILE_ROWS 128
#define ROW_DW    81
#define KCAND     200
#define IOU_THR   0.45f
#define SCORE_THR 0.01f

// ---------------- CDNA5 async global->LDS helpers ----------------
__device__ __forceinline__ void async_copy16(const float* g, float* l) {
#if defined(__gfx1250__) && __has_builtin(__builtin_amdgcn_global_load_async_to_lds_b128)
  // prototype: (global v4i* src, local v4i* dst, imm offset, imm cpol)
  __builtin_amdgcn_global_load_async_to_lds_b128(
      (AS1 v4i*)g, (AS3 v4i*)l, /*offset=*/0, /*cpol=*/0);
#elif defined(__gfx1250__)
  unsigned la = (unsigned)(unsigned long long)(AS3 void*)l;
  unsigned long long ga = (unsigned long long)g;
  asm volatile("global_load_async_to_lds_b128 %0, %1, off"
               :: "v"(la), "v"(ga) : "memory");
#else
  // host-pass / non-gfx1250 fallback
  l[0] = g[0]; l[1] = g[1]; l[2] = g[2]; l[3] = g[3];
#endif
}

__device__ __forceinline__ void wait_async_all() {
#if defined(__gfx1250__) && __has_builtin(__builtin_amdgcn_s_wait_asynccnt)
  __builtin_amdgcn_s_wait_asynccnt(0);
#elif defined(__gfx1250__)
  asm volatile("s_wait_asynccnt 0" ::: "memory");
#endif
}

// ---------------- Stage 1: stream scores, per-anchor max/argmax ----------------
// One block = 128 anchors. Tile = 128*81 floats = 41472 B (always 16B aligned
// since 4 rows = 1296 B = 81*16). Async b128 loads to LDS, then each thread
// reduces its own row (stride-81-dword LDS reads -> conflict-free, 17*t mod 64
// is a permutation over a wave32).
__global__ void k_score(const float* __restrict__ scores, u64* __restrict__ keys, int n) {
  __shared__ __align__(16) float tile[TILE_ROWS * ROW_DW];
  int block0 = blockIdx.x * TILE_ROWS;
  int rows = n - block0;
  if (rows <= 0) return;
  if (rows > TILE_ROWS) rows = TILE_ROWS;

  const float* gbase = scores + (size_t)block0 * ROW_DW;
  int total  = rows * ROW_DW;
  int chunks = total >> 2;                       // 16B chunks
  for (int c = threadIdx.x; c < chunks; c += blockDim.x)
    async_copy16(gbase + 4 * c, &tile[4 * c]);
  // scalar cleanup if total not a multiple of 4 (defensive; 4|rows here)
  for (int r = (chunks << 2) + threadIdx.x; r < total; r += blockDim.x)
    tile[r] = gbase[r];

  wait_async_all();
  __syncthreads();

  int t = threadIdx.x;
  if (t < rows) {
    const float* row = &tile[t * ROW_DW + 1];    // skip background class 0
    float best = row[0];
    int   bl   = 0;
    #pragma unroll 4
    for (int j = 1; j < ROW_DW - 1; ++j) {
      float v = row[j];
      if (v > best) { best = v; bl = j; }        // strict '>' keeps first max (argmax semantics)
    }
    int anchor = block0 + t;
    u64 key = ((u64)__float_as_uint(best) << 32) |
              ((u32)anchor << 8) | (u32)bl;      // score bits | idx(18b) | label(7b)
    keys[anchor] = key;
  }
}

// ---------------- Stage 2: radix-select top-K threshold ----------------
// ctrl[0]=prefix, ctrl[1]=remaining, ctrl[2]=count_gt, ctrl[3]=count_eq
__global__ void k_init(u32* ctrl, u32* hist, const int* cand) {
  hist[threadIdx.x] = 0;
  if (threadIdx.x == 0) {
    ctrl[0] = 0;
    int k = cand[0];
    if (k < 1) k = 1;
    if (k > KCAND) k = KCAND;
    ctrl[1] = (u32)k;
    ctrl[2] = 0;
    ctrl[3] = 0;
  }
}

__global__ void k_zero(u32* hist) { hist[threadIdx.x] = 0; }

__global__ void k_hist(const u64* __restrict__ keys, int n,
                       const u32* __restrict__ ctrl, u32* __restrict__ hist, int shift) {
  __shared__ u32 lh[256];
  lh[threadIdx.x] = 0;
  __syncthreads();
  u32 prefix = ctrl[0];
  u32 mask = (shift >= 24) ? 0u : (0xFFFFFFFFu << (shift + 8));
  int i = blockIdx.x * blockDim.x + threadIdx.x;
  if (i < n) {
    u32 bits = (u32)(keys[i] >> 32);
    if ((bits & mask) == prefix)
      atomicAdd(&lh[(bits >> shift) & 0xFF], 1u);
  }
  __syncthreads();
  if (lh[threadIdx.x]) atomicAdd(&hist[threadIdx.x], lh[threadIdx.x]);
}

__global__ void k_scan(u32* ctrl, const u32* hist, int shift) {
  __shared__ u32 h[256];
  h[threadIdx.x] = hist[threadIdx.x];
  __syncthreads();
  if (threadIdx.x == 0) {
    u32 rem = ctrl[1];
    u32 cum = 0;
    int d = 255;
    for (; d >= 0; --d) {
      u32 c = h[d];
      if (cum + c >= rem) break;                  // k-th largest lives in digit d
      cum += c;
    }
    if (d < 0) d = 0;
    ctrl[0] |= ((u32)d) << shift;
    ctrl[1]  = rem - cum;                         // how many == this digit still needed
  }
}

__global__ void k_compact(const u64* __restrict__ keys, int n, u32* ctrl,
                          u64* __restrict__ gtbuf, u64* __restrict__ eqbuf) {
  int i = blockIdx.x * blockDim.x + threadIdx.x;
  if (i >= n) return;
  u64 key = keys[i];
  u32 bits = (u32)(key >> 32);
  u32 T = ctrl[0];
  if (bits > T) {
    u32 p = atomicAdd(&ctrl[2], 1u);
    if (p < 256) gtbuf[p] = key;
  } else if (bits == T) {
    u32 p = atomicAdd(&ctrl[3], 1u);
    if (p < 256) eqbuf[p] = key;
  }
}

// ---------------- Stage 3: sort + greedy NMS + outputs (one block) ----------------
__global__ void k_final(const float* __restrict__ boxes,
                        const int* wptr, const int* hptr,
                        const u32* __restrict__ ctrl,
                        const u64* __restrict__ gtbuf, const u64* __restrict__ eqbuf,
                        float* __restrict__ out, int K) {
  __shared__ u64 skey[256];
  __shared__ float sx1[256], sy1[256], sx2[256], sy2[256], sarea[256], red[256];
  __shared__ int keep[256];
  int t = threadIdx.x;

  // Merge strictly-greater and tie candidates to exactly K keys (pad with 0).
  u32 G = ctrl[2]; if (G > (u32)K) G = (u32)K;
  u32 avail = ctrl[3]; if (avail > 256u) avail = 256u;
  u32 need = ctrl[1]; if (need > avail) need = avail;
  u64 key = 0;
  if (t < K) {
    if ((u32)t < G) key = gtbuf[t];
    else if ((u32)(t - (int)G) < need) key = eqbuf[t - (int)G];
  }
  skey[t] = key;
  __syncthreads();

  // Bitonic sort, 256 elems, descending (score bits in MSBs dominate).
  for (int kk = 2; kk <= 256; kk <<= 1) {
    for (int j = kk >> 1; j > 0; j >>= 1) {
      int ixj = t ^ j;
      if (ixj > t) {
        u64 a = skey[t], b = skey[ixj];
        bool desc = ((t & kk) == 0);
        if (desc ? (a < b) : (a > b)) { skey[t] = b; skey[ixj] = a; }
      }
      __syncthreads();
    }
  }
  key = skey[t];
  float score = __uint_as_float((u32)(key >> 32));
  u32 lo = (u32)key;
  int idx   = (int)(lo >> 8);
  int label = (int)(lo & 0xFF);

  float bx1 = 0.f, by1 = 0.f, bx2 = 0.f, by2 = 0.f;
  if (t < K) {
    const float* bp = boxes + (size_t)idx * 4;
    bx1 = bp[0]; by1 = bp[1]; bx2 = bp[2]; by2 = bp[3];
  }

  // max over all K boxes' 4 coords (reference: jnp.max(b_top[:, :4]))
  red[t] = (t < K) ? fmaxf(fmaxf(bx1, by1), fmaxf(bx2, by2)) : -3.402823466e38f;
  __syncthreads();
  for (int s = 128; s > 0; s >>= 1) {
    if (t < s) red[t] = fmaxf(red[t], red[t + s]);
    __syncthreads();
  }
  float maxc = red[0];

  float off = (float)label * (maxc + 1.0f);      // per-class coordinate shift
  sx1[t] = bx1 + off; sy1[t] = by1 + off;
  sx2[t] = bx2 + off; sy2[t] = by2 + off;
  sarea[t] = fmaxf(bx2 - bx1, 0.f) * fmaxf(by2 - by1, 0.f);  // shift cancels in area
  keep[t] = (t < K && score > SCORE_THR) ? 1 : 0;
  __syncthreads();

  // Greedy NMS: serial over i, parallel over j>i.
  for (int i = 0; i < K; ++i) {
    if (keep[i] && t > i && t < K && keep[t]) {
      float ix1 = fmaxf(sx1[i], sx1[t]);
      float iy1 = fmaxf(sy1[i], sy1[t]);
      float ix2 = fminf(sx2[i], sx2[t]);
      float iy2 = fminf(sy2[i], sy2[t]);
      float inter = fmaxf(ix2 - ix1, 0.f) * fmaxf(iy2 - iy1, 0.f);
      float uni = sarea[i] + sarea[t] - inter;
      float iou = inter / fmaxf(uni, 1e-12f);
      if (iou > IOU_THR) keep[t] = 0;
    }
    __syncthreads();
  }

  if (t < K) {
    float w = (float)wptr[0], h = (float)hptr[0];
    int kp = keep[t];
    out[t * 4 + 0] = kp ? bx1 * w : 0.f;
    out[t * 4 + 1] = kp ? by1 * h : 0.f;
    out[t * 4 + 2] = kp ? bx2 * w : 0.f;
    out[t * 4 + 3] = kp ? by2 * h : 0.f;
    out[K * 4 + t] = kp ? (float)(label + 1) : 0.f;
    out[K * 5 + t] = kp ? score : 0.f;
  }
}

// ---------------- Host launcher ----------------
extern "C" void kernel_launch(void* const* d_in, const int* in_sizes, int n_in,
                              void* d_out, int out_size, void* d_ws, size_t ws_size,
                              hipStream_t stream) {
  const float* scores  = (const float*)d_in[0];
  const float* boxes   = (const float*)d_in[1];
  const int*   wptr    = (const int*)d_in[2];
  const int*   hptr    = (const int*)d_in[3];
  const int*   candptr = (const int*)d_in[4];
  float* out = (float*)d_out;

  int n = in_sizes[1] / 4;                       // anchor count from boxes shape

  // Workspace layout (re-initialized every call; ~2.01 MB total):
  char* ws = (char*)d_ws;
  u32* ctrl  = (u32*)ws;          // 16 u32 control words
  u32* hist  = (u32*)(ws + 64);   // 256-bin radix histogram
  u64* gtbuf = (u64*)(ws + 2048); // strictly-greater candidates
  u64* eqbuf = (u64*)(ws + 4096); // tie candidates
  u64* keys  = (u64*)(ws + 8192); // n packed keys

  int sblocks = (n + TILE_ROWS - 1) / TILE_ROWS;
  k_score<<<sblocks, TILE_ROWS, 0, stream>>>(scores, keys, n);

  int hblocks = (n + 255) / 256;
  k_init<<<1, 256, 0, stream>>>(ctrl, hist, candptr);
  for (int p = 0; p < 4; ++p) {
    int shift = 24 - 8 * p;
    if (p > 0) k_zero<<<1, 256, 0, stream>>>(hist);
    k_hist<<<hblocks, 256, 0, stream>>>(keys, n, ctrl, hist, shift);
    k_scan<<<1, 256, 0, stream>>>(ctrl, hist, shift);
  }
  k_compact<<<hblocks, 256, 0, stream>>>(keys, n, ctrl, gtbuf, eqbuf);
  k_final<<<1, 256, 0, stream>>>(boxes, wptr, hptr, ctrl, gtbuf, eqbuf, out, KCAND);
}